// SurfEval_70317204570141
// MI455X (gfx1250) — compile-verified
//
#include <hip/hip_runtime.h>

// MI455X (gfx1250) NURBS surface evaluation.
// Banded basis contraction mapped onto V_WMMA_F32_16X16X4_F32 (wave32).
//
// Roofline: 50.3 MB output @ 23.3 TB/s => ~2.2us floor. Banded formulation is
// ~340 MFLOP total => bandwidth-bound. We tile the band: any 16 consecutive
// v-samples span <= 6 contiguous control columns (cubic spline, 61 segments
// over 512 samples), so each 16x16 output tile is an exact dense GEMM with
// K=8 (zero padded) -> 2 wmma k-steps x 4 channels per tile.
//
// Round 1 -> 2 changes:
//  * Epilogue divide: replace IEEE div expansion (~25 VALU/divide, would be
//    the bottleneck at ~3.4us of pure VALU) with v_rcp_f32 + one NR step
//    (3 VALU, ~0.5 ulp; w is a positive weight blend in [0.5,1.5]).
//  * Output stores are non-temporal (write-once 50MB stream, never re-read).

typedef __attribute__((ext_vector_type(2))) float v2f;
typedef __attribute__((ext_vector_type(8))) float v8f;

#define NCTRL 64
#define OUTN  512
#define KTILE 8     // padded band width: two 16x16x4 k-steps

// ---------------------------------------------------------------------------
// Stage 1: A[b][c][u][n] = sum_r Nu[u][r] * ctrl[b][uspan[u]-3+r][n][c]
// 2M elements, 4 FMAs each; memory bound (writes 8 MiB to d_ws).
// Layout [b][c][u][n] puts the WMMA K dim (n) innermost for stage 2.
// ---------------------------------------------------------------------------
__global__ void nurbs_stage1(const float* __restrict__ ctrl,
                             const int*   __restrict__ uspan,
                             const float* __restrict__ Nu,
                             float*       __restrict__ A)
{
    int idx = blockIdx.x * blockDim.x + threadIdx.x; // ((b*4+c)*512+u)*64+n
    int n = idx & 63;
    int u = (idx >> 6) & 511;
    int c = (idx >> 15) & 3;
    int b = idx >> 17;

    int m0 = uspan[u] - 3;                 // span in [3,63] -> m0 in [0,60]
    float acc = 0.0f;
#pragma unroll
    for (int r = 0; r < 4; ++r) {
        float w = Nu[u * 4 + r];
        float x = ctrl[(((size_t)b * 64 + (m0 + r)) * 64 + n) * 4 + c];
        acc = fmaf(w, x, acc);
    }
    A[idx] = acc;
}

// ---------------------------------------------------------------------------
// Stage 2: one wave per (b, 16u x 16v) tile.
//   S[u][v] = sum_k Aband[u][k] * Nvd[k][v],  k = n - kbase, kbase=vspan[v0]-3
// Nvd is the 8x16 densified band slice of Nv, built in LDS.
// Four f32 channels accumulate via V_WMMA_F32_16X16X4_F32; epilogue divides
// xyz by w (fast rcp + 1 NR) and streams 16x16x3 coalesced NT floats.
// ---------------------------------------------------------------------------
__global__ void nurbs_stage2(const float* __restrict__ A,
                             const int*   __restrict__ vspan,
                             const float* __restrict__ Nv,
                             float*       __restrict__ out)
{
    __shared__ float nvd[8][KTILE * 16];   // per-wave dense band tile

    const int lane = threadIdx.x & 31;
    const int wave = threadIdx.x >> 5;
    const int tile = blockIdx.x * 8 + wave;        // 16384 tiles total
    const int vt = tile & 31;
    const int ut = (tile >> 5) & 31;
    const int b  = tile >> 10;
    const int u0 = ut * 16;
    const int v0 = vt * 16;

    const int kbase = vspan[v0] - 3;               // broadcast load

    // Densify this tile's slice of Nv: nvd[k][v] (zero-padded band).
    if (lane < 16) {
#pragma unroll
        for (int k = 0; k < KTILE; ++k) nvd[wave][k * 16 + lane] = 0.0f;
        int v  = v0 + lane;
        int ki = vspan[v] - 3 - kbase;             // 0..2 for this problem
#pragma unroll
        for (int s = 0; s < 4; ++s) {
            int kk = ki + s;
            if (kk < KTILE) nvd[wave][kk * 16 + lane] = Nv[v * 4 + s];
        }
    }
    __syncthreads();

    const int mrow  = lane & 15;        // M (=u) for A-operand, N (=v) for B
    const int khalf = (lane >> 4) * 2;  // per-half-wave K pair (ISA A layout)

    v8f acc[4];
#pragma unroll
    for (int c = 0; c < 4; ++c) {
        v8f d = {};
#pragma unroll
        for (int ks = 0; ks < 2; ++ks) {
            int kk = ks * 4 + khalf;
            // Per-element clamp: any n>63 slot has nvd==0 (beyond the band),
            // so the clamped (finite) value contributes exactly zero.
            int n0 = kbase + kk;     if (n0 > 63) n0 = 63;
            int n1 = kbase + kk + 1; if (n1 > 63) n1 = 63;
            const float* ap =
                A + (((size_t)(b * 4 + c) * OUTN + (u0 + mrow)) * NCTRL);
            v2f av; av.x = ap[n0]; av.y = ap[n1];
            v2f bv;
            bv.x = nvd[wave][kk * 16 + mrow];
            bv.y = nvd[wave][(kk + 1) * 16 + mrow];
            d = __builtin_amdgcn_wmma_f32_16x16x4_f32(
                    /*neg_a=*/false, av, /*neg_b=*/false, bv,
                    /*c_mod=*/(short)0, d, /*reuse_a=*/false, /*reuse_b=*/false);
        }
        acc[c] = d;
    }

    // Epilogue: C/D layout -> VGPR j holds (M = j + 8*(lane>>4), N = lane&15).
#pragma unroll
    for (int j = 0; j < 8; ++j) {
        int u = u0 + j + (lane >> 4) * 8;
        int v = v0 + (lane & 15);
        float w = acc[3][j];
        // Fast reciprocal: v_rcp_f32 + one Newton-Raphson refinement.
        float r = __builtin_amdgcn_rcpf(w);
        r = r * (2.0f - w * r);
        size_t o = (((size_t)b * OUTN + u) * OUTN + v) * 3;
        __builtin_nontemporal_store(acc[0][j] * r, &out[o + 0]);
        __builtin_nontemporal_store(acc[1][j] * r, &out[o + 1]);
        __builtin_nontemporal_store(acc[2][j] * r, &out[o + 2]);
    }
}

// ---------------------------------------------------------------------------
extern "C" void kernel_launch(void* const* d_in, const int* in_sizes, int n_in,
                              void* d_out, int out_size, void* d_ws, size_t ws_size,
                              hipStream_t stream)
{
    const float* ctrl  = (const float*)d_in[0];   // (16,64,64,4) f32
    const int*   uspan = (const int*)  d_in[1];   // (512,) i32
    const int*   vspan = (const int*)  d_in[2];   // (512,) i32
    const float* Nu    = (const float*)d_in[3];   // (512,4) f32
    const float* Nv    = (const float*)d_in[4];   // (512,4) f32
    float*       out   = (float*)d_out;           // (16,512,512,3) f32
    float*       A     = (float*)d_ws;            // needs 16*4*512*64*4 B = 8 MiB

    // Stage 1: 16*4*512*64 = 2,097,152 elements
    nurbs_stage1<<<8192, 256, 0, stream>>>(ctrl, uspan, Nu, A);
    // Stage 2: 16*32*32 = 16384 tiles, 8 waves (tiles) per block
    nurbs_stage2<<<2048, 256, 0, stream>>>(A, vspan, Nv, out);
}